// clustering_attention_dynamic_learning1_45286135169494
// MI455X (gfx1250) — compile-verified
//
#include <hip/hip_runtime.h>
#include <hip/hip_bf16.h>
#include <math.h>

// Problem constants (B, N, K, S_IN=S_OUT, C, 4*S_OUT)
#define BB   32
#define NN   325
#define KK   20
#define SS   12
#define CC   10
#define HID  48
#define OUT_SCAL (BB * NN * CC * SS)   // 1,248,000 : [cluster_loss, dist_mean, wh_mean]

typedef float v2f __attribute__((ext_vector_type(2)));
typedef float v8f __attribute__((ext_vector_type(8)));

__device__ __forceinline__ float leaky(float x) { return x >= 0.0f ? x : 0.5f * x; }

__global__ void init_scalars_kernel(float* __restrict__ out) {
  out[OUT_SCAL + 0] = 0.0f;
  out[OUT_SCAL + 1] = 0.0f;
  out[OUT_SCAL + 2] = 0.0f;
}

// One wave32 block per (b, n). The entire gathered-attention pipeline runs on
// V_WMMA_F32_16X16X4_F32 tiles (fp32-exact):
//   Whm = leaky(Xm @ w_W + w_b)               6 wmma
//   H   = leaky(Whm @ a1_W[12:] + hx + a1_b) 18 wmma
//   Att = leaky(H @ a2_W + a2_b)             24 wmma
//   out = Am^T @ Wht                          5 wmma
//   dist = Wtn Wtn^T, prob = Am Am^T         24 wmma
__global__ __launch_bounds__(32) void fused_attn_kernel(
    const float* __restrict__ input_data,   // (B,N,12)
    const float* __restrict__ w_W,          // (12,12)
    const float* __restrict__ w_b,          // (12)
    const float* __restrict__ a1_W,         // (24,48)
    const float* __restrict__ a1_b,         // (48)
    const float* __restrict__ a2_W,         // (48,10)
    const float* __restrict__ a2_b,         // (10)
    const int*   __restrict__ topk,         // (B,N,20)
    float* __restrict__ out)                // (B,N,10,12) ++ 3 scalars
{
  __shared__ float sWWp[12][16];   // w_W padded (cols >=12 zero)
  __shared__ float sA2p[48][16];   // a2_W padded (cols >=10 zero)
  __shared__ float sXm [32][12];   // gathered input rows (rows >=20 zero)
  __shared__ float sWht[32][16];   // wh_topk (rows >=20, cols >=12 zero)
  __shared__ float sH  [32][48];   // hidden (rows >=20 zero)
  __shared__ float sAtt[32][16];   // attention pre-softmax; reused for out tile
  __shared__ float sAm [32][16];   // softmax attention (rows >=20, cols >=10 zero)
  __shared__ float sWtn[32][12];   // normalized wh_topk
  __shared__ float sWhn[SS];
  __shared__ float sHx [HID];

  const int bn   = blockIdx.x;
  const int b    = bn / NN;
  const int lane = threadIdx.x;
  const int ll   = lane & 15;
  const int hi   = lane >> 4;
  const int koff = hi << 1;        // WMMA A/B K-split across lane halves

  // ---- stage zero-padded weight tiles ----
  for (int t = lane; t < 12 * 16; t += 32) {
    const int r = t >> 4, c = t & 15;
    sWWp[r][c] = (c < SS) ? w_W[r * SS + c] : 0.0f;
  }
  for (int t = lane; t < 48 * 16; t += 32) {
    const int r = t >> 4, c = t & 15;
    sA2p[r][c] = (c < CC) ? a2_W[r * CC + c] : 0.0f;
  }

  // ---- wh[b,n,:] (lanes 0..11) and hx = wh_n @ a1_W[:12] ----
  const float* xn = input_data + bn * SS;
  if (lane < SS) {
    float acc = w_b[lane];
    #pragma unroll
    for (int i = 0; i < SS; ++i) acc += xn[i] * w_W[i * SS + lane];
    sWhn[lane] = leaky(acc);
  }
  __syncthreads();
  for (int j = lane; j < HID; j += 32) {
    float acc = 0.0f;
    #pragma unroll
    for (int s = 0; s < SS; ++s) acc += sWhn[s] * a1_W[s * HID + j];
    sHx[j] = acc;
  }

  // ---- gather neighbor input rows into sXm ----
  const bool active = lane < KK;
  const int  m      = active ? topk[bn * KK + lane] : 0;
  const float* xm   = input_data + (b * NN + m) * SS;
  #pragma unroll
  for (int i = 0; i < SS; ++i) sXm[lane][i] = active ? xm[i] : 0.0f;
  __syncthreads();

  const float bias_wb = (ll < SS) ? w_b[ll]  : 0.0f;
  const float bias_a2 = (ll < CC) ? a2_b[ll] : 0.0f;

  // ================= WMMA step 1: Whm = leaky(Xm @ w_W + w_b) ==============
  #pragma unroll
  for (int Mt = 0; Mt < 32; Mt += 16) {
    v8f acc = {};
    #pragma unroll
    for (int k0 = 0; k0 < 12; k0 += 4) {
      v2f a, bf;
      a[0]  = sXm [Mt + ll][k0 + koff];  a[1]  = sXm [Mt + ll][k0 + koff + 1];
      bf[0] = sWWp[k0 + koff][ll];       bf[1] = sWWp[k0 + koff + 1][ll];
      acc = __builtin_amdgcn_wmma_f32_16x16x4_f32(false, a, false, bf,
                                                  (short)0, acc, false, false);
    }
    #pragma unroll
    for (int j = 0; j < 8; ++j) {
      const int r = Mt + j + 8 * hi;
      sWht[r][ll] = (r < KK && ll < SS) ? leaky(acc[j] + bias_wb) : 0.0f;
    }
  }
  __syncthreads();

  // ===== WMMA step 2: H = leaky(Whm @ a1_W[12:] + hx + a1_b) ===============
  #pragma unroll
  for (int Nt = 0; Nt < HID; Nt += 16) {
    const int jj = Nt + ll;
    const float add_j = sHx[jj] + a1_b[jj];
    #pragma unroll
    for (int Mt = 0; Mt < 32; Mt += 16) {
      v8f acc = {};
      #pragma unroll
      for (int k0 = 0; k0 < 12; k0 += 4) {
        v2f a, bf;
        a[0]  = sWht[Mt + ll][k0 + koff];  a[1]  = sWht[Mt + ll][k0 + koff + 1];
        bf[0] = a1_W[(SS + k0 + koff) * HID + jj];
        bf[1] = a1_W[(SS + k0 + koff + 1) * HID + jj];
        acc = __builtin_amdgcn_wmma_f32_16x16x4_f32(false, a, false, bf,
                                                    (short)0, acc, false, false);
      }
      #pragma unroll
      for (int j = 0; j < 8; ++j) {
        const int r = Mt + j + 8 * hi;
        sH[r][jj] = (r < KK) ? leaky(acc[j] + add_j) : 0.0f;
      }
    }
  }
  __syncthreads();

  // ===== WMMA step 3: Att = leaky(H @ a2_W + a2_b) =========================
  #pragma unroll
  for (int Mt = 0; Mt < 32; Mt += 16) {
    v8f acc = {};
    #pragma unroll
    for (int k0 = 0; k0 < HID; k0 += 4) {
      v2f a, bf;
      a[0]  = sH  [Mt + ll][k0 + koff];  a[1]  = sH  [Mt + ll][k0 + koff + 1];
      bf[0] = sA2p[k0 + koff][ll];       bf[1] = sA2p[k0 + koff + 1][ll];
      acc = __builtin_amdgcn_wmma_f32_16x16x4_f32(false, a, false, bf,
                                                  (short)0, acc, false, false);
    }
    #pragma unroll
    for (int j = 0; j < 8; ++j) {
      const int r = Mt + j + 8 * hi;
      sAtt[r][ll] = (r < KK && ll < CC) ? leaky(acc[j] + bias_a2) : 0.0f;
    }
  }
  __syncthreads();

  // ---- per-lane softmax over C, and normalized wh_topk ----
  float att[CC];
  #pragma unroll
  for (int c = 0; c < CC; ++c) att[c] = sAtt[lane][c];
  float mx = att[0];
  #pragma unroll
  for (int c = 1; c < CC; ++c) mx = fmaxf(mx, att[c]);
  float esum = 0.0f;
  float e[CC];
  #pragma unroll
  for (int c = 0; c < CC; ++c) { e[c] = expf(att[c] - mx); esum += e[c]; }
  const float inv_esum = 1.0f / esum;

  float wv[SS];
  float nrm2 = 0.0f;
  #pragma unroll
  for (int s = 0; s < SS; ++s) { wv[s] = sWht[lane][s]; nrm2 += wv[s] * wv[s]; }
  const float invn = 1.0f / (sqrtf(nrm2) + 1e-8f);

  #pragma unroll
  for (int s = 0; s < SS; ++s) sWtn[lane][s] = wv[s] * invn;          // rows>=20 stay 0
  #pragma unroll
  for (int c = 0; c < CC; ++c) sAm[lane][c] = active ? e[c] * inv_esum : 0.0f;
  #pragma unroll
  for (int c = CC; c < 16; ++c) sAm[lane][c] = 0.0f;
  __syncthreads();

  // ===== WMMA step 4: output[c,s] = sum_k Am[k,c] Wht[k,s]  (K=20) =========
  v8f accO = {};
  #pragma unroll
  for (int k0 = 0; k0 < KK; k0 += 4) {
    v2f a, bf;
    a[0]  = sAm [k0 + koff][ll];  a[1]  = sAm [k0 + koff + 1][ll];
    bf[0] = sWht[k0 + koff][ll];  bf[1] = sWht[k0 + koff + 1][ll];
    accO = __builtin_amdgcn_wmma_f32_16x16x4_f32(false, a, false, bf,
                                                 (short)0, accO, false, false);
  }
  __syncthreads();
  #pragma unroll
  for (int j = 0; j < 8; ++j) sAtt[j + 8 * hi][ll] = accO[j];  // stage D tile
  __syncthreads();
  for (int t = lane; t < CC * SS; t += 32) {                   // coalesced store
    const int c = t / SS, s = t - c * SS;
    out[bn * (CC * SS) + t] = sAtt[c][s];
  }

  // ===== WMMA step 5: dist = Wtn Wtn^T, prob = Am Am^T (20x20, K=12) =======
  float distsum = 0.0f, clsum = 0.0f;
  #pragma unroll
  for (int Mt = 0; Mt < 32; Mt += 16) {
    #pragma unroll
    for (int Nt = 0; Nt < 32; Nt += 16) {
      v8f accD = {};
      v8f accP = {};
      #pragma unroll
      for (int k0 = 0; k0 < 12; k0 += 4) {
        v2f ad, bd, ap, bp;
        ad[0] = sWtn[Mt + ll][k0 + koff];  ad[1] = sWtn[Mt + ll][k0 + koff + 1];
        bd[0] = sWtn[Nt + ll][k0 + koff];  bd[1] = sWtn[Nt + ll][k0 + koff + 1];
        accD = __builtin_amdgcn_wmma_f32_16x16x4_f32(false, ad, false, bd,
                                                     (short)0, accD, false, false);
        ap[0] = sAm[Mt + ll][k0 + koff];   ap[1] = sAm[Mt + ll][k0 + koff + 1];
        bp[0] = sAm[Nt + ll][k0 + koff];   bp[1] = sAm[Nt + ll][k0 + koff + 1];
        accP = __builtin_amdgcn_wmma_f32_16x16x4_f32(false, ap, false, bp,
                                                     (short)0, accP, false, false);
      }
      #pragma unroll
      for (int j = 0; j < 8; ++j) {
        const int r = Mt + j + 8 * hi;
        const int c = Nt + ll;
        if (r < KK && c < KK) {
          const float d = accD[j];
          distsum += d;
          if (r != c) {
            const float p  = fminf(fmaxf(accP[j], 1e-4f), 1.0f - 1e-4f);
            const float lp = logf(p);
            clsum += (d >= 0.5f) ? -lp : lp;   // cl = (diff - same) * log p
          }
        }
      }
    }
  }

  // ---- wave reduction + scalar accumulation ----
  #pragma unroll
  for (int off = 16; off > 0; off >>= 1) {
    distsum += __shfl_xor(distsum, off, 32);
    clsum   += __shfl_xor(clsum,   off, 32);
  }
  if (lane == 0) {
    atomicAdd(out + OUT_SCAL + 0, clsum   * (1.0f / (float)(BB * NN)));
    atomicAdd(out + OUT_SCAL + 1, distsum * (1.0f / ((float)BB * NN * KK * KK)));
    float ws = 0.0f;
    #pragma unroll
    for (int s = 0; s < SS; ++s) ws += sWhn[s];
    atomicAdd(out + OUT_SCAL + 2, ws * (1.0f / ((float)BB * NN * SS)));
  }
}

extern "C" void kernel_launch(void* const* d_in, const int* in_sizes, int n_in,
                              void* d_out, int out_size, void* d_ws, size_t ws_size,
                              hipStream_t stream) {
  (void)in_sizes; (void)n_in; (void)out_size; (void)d_ws; (void)ws_size;
  // setup_inputs order: 0 fushed_features (unused), 1 input_data, 2 w_W, 3 w_b,
  //                     4 a1_W, 5 a1_b, 6 a2_W, 7 a2_b, 8 adj_mx_topk_index
  const float* input_data = (const float*)d_in[1];
  const float* w_W  = (const float*)d_in[2];
  const float* w_b  = (const float*)d_in[3];
  const float* a1_W = (const float*)d_in[4];
  const float* a1_b = (const float*)d_in[5];
  const float* a2_W = (const float*)d_in[6];
  const float* a2_b = (const float*)d_in[7];
  const int*   topk = (const int*)d_in[8];
  float* out = (float*)d_out;

  init_scalars_kernel<<<1, 1, 0, stream>>>(out);
  fused_attn_kernel<<<BB * NN, 32, 0, stream>>>(input_data, w_W, w_b,
                                                a1_W, a1_b, a2_W, a2_b,
                                                topk, out);
}